// VMambaBlock_65189013618744
// MI455X (gfx1250) — compile-verified
//
#include <hip/hip_runtime.h>

// ---------------------------------------------------------------------------
// VMamba block for MI455X (gfx1250): fp32 WMMA GEMMs + fused selective scan.
// Shapes fixed by the reference: BB=8 base batch, 4 directions, L=1024,
// C=256, I=512, N=R=16.
// ---------------------------------------------------------------------------

#define BB 8
#define ND 4
#define LL 1024
#define CC 256
#define II 512
#define NS 16
#define RR 16
#define EPSF 1e-5f

typedef __attribute__((ext_vector_type(2))) float v2f;
typedef __attribute__((ext_vector_type(8))) float v8f;

// Direction permutation (involution): d bit0 flips W, bit1 flips H.
__device__ __forceinline__ int perm_pos(int d, int l) {
  int h = l >> 5, w = l & 31;
  if (d & 1) w = 31 - w;
  if (d & 2) h = 31 - h;
  return (h << 5) | w;
}

// ---------------------------------------------------------------------------
// 1) rmsnorm over rows of 256 (one block per row, wave32-safe LDS reduce)
// ---------------------------------------------------------------------------
__global__ __launch_bounds__(CC) void rmsnorm_kernel(
    const float* __restrict__ x, const float* __restrict__ w,
    float* __restrict__ out) {
  __shared__ float red[CC];
  const int row = blockIdx.x, c = threadIdx.x;
  float v = x[(size_t)row * CC + c];
  red[c] = v * v;
  __syncthreads();
  for (int s = CC / 2; s > 0; s >>= 1) {
    if (c < s) red[c] += red[c + s];
    __syncthreads();
  }
  float sc = rsqrtf(red[0] * (1.0f / CC) + EPSF);
  out[(size_t)row * CC + c] = v * sc * w[c];
}

// ---------------------------------------------------------------------------
// 2) fp32 WMMA GEMM: C[m,n] = sum_k A[m,k] * Bw[n,k]
//    A:[M,K] row-major activations, Bw:[N,K] row-major weight (i.e. A @ Bw^T).
//    One wave per 16x16 output tile, K-loop of V_WMMA_F32_16X16X4_F32.
//    fp32 A layout: lane L holds rows m=L%16, k-pair base 2*(L/16).
//    fp32 B layout: lane L holds cols n=L%16, same k-pair.
//    fp32 D layout: lane L col n=L%16, row r + 8*(L/16) for vgpr r.
// ---------------------------------------------------------------------------
__global__ __launch_bounds__(32) void wmma_gemm_f32(
    const float* __restrict__ A, const float* __restrict__ Bw,
    float* __restrict__ C, int M, int N, int K) {
  const int tm = blockIdx.x, tn = blockIdx.y;
  const int lane = threadIdx.x;
  const int half = lane >> 4;   // 0 | 1
  const int lm = lane & 15;
  const float* __restrict__ Ar = A + (size_t)(tm * 16 + lm) * K + 2 * half;
  const float* __restrict__ Br = Bw + (size_t)(tn * 16 + lm) * K + 2 * half;
  v8f acc = {};
  for (int k = 0; k < K; k += 4) {
    v2f a, b;
    a.x = Ar[k];
    a.y = Ar[k + 1];
    b.x = Br[k];
    b.y = Br[k + 1];
    acc = __builtin_amdgcn_wmma_f32_16x16x4_f32(false, a, false, b, (short)0,
                                                acc, false, false);
  }
  float* __restrict__ Co = C + (size_t)(tm * 16 + 8 * half) * N + tn * 16 + lm;
#pragma unroll
  for (int r = 0; r < 8; ++r) Co[(size_t)r * N] = acc[r];
}

// ---------------------------------------------------------------------------
// 3) causal depthwise conv (k=4) + bias + SiLU, reading the base-order in-proj
//    through the direction permutation. One block per (dir,batch,l), 512 lanes.
// ---------------------------------------------------------------------------
__global__ __launch_bounds__(II) void conv_silu_kernel(
    const float* __restrict__ proj, const float* __restrict__ conv_w,
    const float* __restrict__ conv_b, float* __restrict__ hs) {
  const int db = blockIdx.x >> 10;  // d*BB + b
  const int l = blockIdx.x & (LL - 1);
  const int d = db >> 3, b = db & 7;
  const int i = threadIdx.x;
  float acc = conv_b[i];
#pragma unroll
  for (int k = 0; k < 4; ++k) {
    int ls = l - 3 + k;
    if (ls >= 0) {
      int p = perm_pos(d, ls);
      acc += conv_w[i * 4 + k] * proj[((size_t)b * LL + p) * (2 * II) + i];
    }
  }
  float s = acc / (1.0f + __expf(-acc));  // silu
  hs[((size_t)db * LL + l) * II + i] = s;
}

// ---------------------------------------------------------------------------
// 4) fused selective scan: per lane = one (dir*batch, channel i).
//    16 SSM states in registers; dt-GEMV + softplus fused; output
//    y = (scan + u*Dp) * silu(gate) written in-place over hs.
// ---------------------------------------------------------------------------
__global__ __launch_bounds__(128) void scan_kernel(
    float* hs_yg /* in: hs, out: gated y (in-place) */,
    const float* __restrict__ ssm, const float* __restrict__ proj,
    const float* __restrict__ W_dt, const float* __restrict__ b_dt,
    const float* __restrict__ A_log, const float* __restrict__ Dp) {
  const int db = blockIdx.y;  // 0..31
  const int d = db >> 3, b = db & 7;
  const int i = blockIdx.x * 128 + threadIdx.x;  // channel 0..511
  __shared__ float sm[RR + 2 * NS];              // dt_r | B | C for current l

  float Arow[NS], Wdt[RR], st[NS];
#pragma unroll
  for (int n = 0; n < NS; ++n) {
    Arow[n] = -__expf(A_log[(size_t)i * NS + n]);
    Wdt[n] = W_dt[(size_t)i * RR + n];
    st[n] = 0.0f;
  }
  const float bdt = b_dt[i], dpi = Dp[i];
  const float* __restrict__ ssmb = ssm + (size_t)db * LL * (RR + 2 * NS);
  float* hsb = hs_yg + (size_t)db * LL * II;

  for (int l = 0; l < LL; ++l) {
    if (threadIdx.x < RR + 2 * NS)
      sm[threadIdx.x] = ssmb[(size_t)l * (RR + 2 * NS) + threadIdx.x];
    __syncthreads();

    float u = hsb[(size_t)l * II + i];
    int p = perm_pos(d, l);
    float g = proj[((size_t)b * LL + p) * (2 * II) + II + i];
    if (l + 4 < LL) __builtin_prefetch(&hsb[(size_t)(l + 4) * II + i], 0, 1);

    float z = bdt;
#pragma unroll
    for (int r = 0; r < RR; ++r) z += sm[r] * Wdt[r];
    float dt = (z > 20.0f) ? z : log1pf(__expf(z));  // softplus
    float du = dt * u;

    float y = 0.0f;
#pragma unroll
    for (int n = 0; n < NS; ++n) {
      float dA = __expf(dt * Arow[n]);
      st[n] = dA * st[n] + du * sm[RR + n];
      y += st[n] * sm[RR + NS + n];
    }
    y += u * dpi;
    y *= g / (1.0f + __expf(-g));  // * silu(gate)
    __syncthreads();               // sm consumed before next overwrite
    hsb[(size_t)l * II + i] = y;   // in-place: u already consumed
  }
}

// ---------------------------------------------------------------------------
// 5) combine: per (b,p) row -> for each direction: residual + rmsnorm(rms_wf),
//    un-flip, average, then final layernorm. One block per row.
// ---------------------------------------------------------------------------
__global__ __launch_bounds__(CC) void combine_kernel(
    const float* __restrict__ x, const float* __restrict__ mixout,
    const float* __restrict__ rms_wf, const float* __restrict__ ln_w,
    const float* __restrict__ ln_b, float* __restrict__ out) {
  __shared__ float red[CC];
  const int b = blockIdx.x >> 10;
  const int p = blockIdx.x & (LL - 1);
  const int c = threadIdx.x;
  const float xv = x[((size_t)b * LL + p) * CC + c];
  float acc = 0.0f;
#pragma unroll
  for (int d = 0; d < ND; ++d) {
    int l = perm_pos(d, p);  // involution: direction-d contribution to p
    float v = xv + mixout[(((size_t)(d * BB + b)) * LL + l) * CC + c];
    red[c] = v * v;
    __syncthreads();
    for (int s = CC / 2; s > 0; s >>= 1) {
      if (c < s) red[c] += red[c + s];
      __syncthreads();
    }
    float sc = rsqrtf(red[0] * (1.0f / CC) + EPSF);
    __syncthreads();
    acc += v * sc * rms_wf[c];
  }
  acc *= 0.25f;
  // layernorm
  red[c] = acc;
  __syncthreads();
  for (int s = CC / 2; s > 0; s >>= 1) {
    if (c < s) red[c] += red[c + s];
    __syncthreads();
  }
  float m = red[0] * (1.0f / CC);
  __syncthreads();
  float dv = acc - m;
  red[c] = dv * dv;
  __syncthreads();
  for (int s = CC / 2; s > 0; s >>= 1) {
    if (c < s) red[c] += red[c + s];
    __syncthreads();
  }
  float var = red[0] * (1.0f / CC);
  out[((size_t)b * LL + p) * CC + c] =
      dv * rsqrtf(var + EPSF) * ln_w[c] + ln_b[c];
}

// ---------------------------------------------------------------------------
extern "C" void kernel_launch(void* const* d_in, const int* in_sizes, int n_in,
                              void* d_out, int out_size, void* d_ws,
                              size_t ws_size, hipStream_t stream) {
  const float* x = (const float*)d_in[0];       // [8,32,32,256]
  const float* W_in = (const float*)d_in[1];    // [1024,256]
  const float* conv_w = (const float*)d_in[2];  // [512,1,4]
  const float* conv_b = (const float*)d_in[3];  // [512]
  const float* W_x = (const float*)d_in[4];     // [48,512]
  const float* W_dt = (const float*)d_in[5];    // [512,16]
  const float* b_dt = (const float*)d_in[6];    // [512]
  const float* A_log = (const float*)d_in[7];   // [512,16]
  const float* Dp = (const float*)d_in[8];      // [512]
  const float* W_out = (const float*)d_in[9];   // [256,512]
  const float* rms_w1 = (const float*)d_in[10];
  const float* rms_wf = (const float*)d_in[11];
  const float* ln_w = (const float*)d_in[12];
  const float* ln_b = (const float*)d_in[13];
  float* out = (float*)d_out;

  // Workspace layout (all fits in the 192MB L2):
  char* ws = (char*)d_ws;
  float* xn = (float*)(ws);                          //  8 MiB [8192,256]
  float* proj = (float*)(ws + (8ull << 20));         // 32 MiB [8192,1024]
  float* hs = (float*)(ws + (40ull << 20));          // 64 MiB [32,1024,512]
  float* ssm = (float*)(ws + (104ull << 20));        //  6 MiB [32,1024,48]
  float* mixout = (float*)(ws + (110ull << 20));     // 32 MiB [32,1024,256]

  // 1) rmsnorm(x) * rms_w1 (base order; norm commutes with the flips)
  rmsnorm_kernel<<<BB * LL, CC, 0, stream>>>(x, rms_w1, xn);

  // 2) in-proj once on base batch: proj = xn @ W_in^T  [8192,1024]
  wmma_gemm_f32<<<dim3((BB * LL) / 16, (2 * II) / 16), 32, 0, stream>>>(
      xn, W_in, proj, BB * LL, 2 * II, CC);

  // 3) permuted causal conv + SiLU -> hs [32,1024,512]
  conv_silu_kernel<<<ND * BB * LL, II, 0, stream>>>(proj, conv_w, conv_b, hs);

  // 4) x-proj: ssm = hs @ W_x^T  [32768,48]
  wmma_gemm_f32<<<dim3((ND * BB * LL) / 16, (RR + 2 * NS) / 16), 32, 0,
                 stream>>>(hs, W_x, ssm, ND * BB * LL, RR + 2 * NS, II);

  // 5) fused selective scan (dt-GEMV + softplus + scan + D-skip + gate SiLU),
  //    writes gated y in-place over hs
  scan_kernel<<<dim3(II / 128, ND * BB), 128, 0, stream>>>(
      hs, ssm, proj, W_dt, b_dt, A_log, Dp);

  // 6) out-proj: mixout = yg @ W_out^T  [32768,256]
  wmma_gemm_f32<<<dim3((ND * BB * LL) / 16, CC / 16), 32, 0, stream>>>(
      hs, W_out, mixout, ND * BB * LL, CC, II);

  // 7) residual + per-direction rmsnorm + un-flip average + layernorm
  combine_kernel<<<BB * LL, CC, 0, stream>>>(x, mixout, rms_wf, ln_w, ln_b,
                                             out);
}